// MultiModalAttentionGraph_20744692040131
// MI455X (gfx1250) — compile-verified
//
#include <hip/hip_runtime.h>
#include <hip/hip_bf16.h>
#include <cstdint>

// MI455X / gfx1250, wave32. fp32 WMMA path: V_WMMA_F32_16X16X4_F32.
// N=8, L=1024, E=512, H=4. out = [2048, 2048] f32.

typedef __attribute__((ext_vector_type(2))) float v2f;
typedef __attribute__((ext_vector_type(8))) float v8f;

#define NB   8
#define LL   1024
#define EE   512
#define PROJ_ELEMS ((size_t)NB * LL * EE)   // 4,194,304 floats = 16 MB per matrix

// inv_scale = 1 / (H * sqrt(E)) = 1 / (4 * sqrt(512))
#define INV_SCALE 0.011048543456039805f
#define LOG2E     1.4426950408889634f

// ---------------------------------------------------------------------------
// Kernel 1: fused 4-way projection  Y = X @ W^T + b  (fp32 WMMA 16x16x4)
// grid: (8192/64 row blocks, 512/64 col blocks, 4 projections), block: 128 thr
// Each wave computes a 16x64 strip: 4 accumulator tiles, K-loop step 4.
// A-frag (16x4): lane<16 holds K={k,k+1}, lane>=16 holds K={k+2,k+3}, row=lane&15
//   -> contiguous float2 at X[row][k + 2*hi].
// B-frag (4x16) = W^T tile: element (K,N) = W[N][K]
//   -> contiguous float2 at W[col][k + 2*hi].
// ---------------------------------------------------------------------------
__global__ __launch_bounds__(128) void proj_kernel(
    const float* __restrict__ modalA, const float* __restrict__ modalB,
    const float* __restrict__ WqA, const float* __restrict__ bqA,
    const float* __restrict__ WkA, const float* __restrict__ bkA,
    const float* __restrict__ WqB, const float* __restrict__ bqB,
    const float* __restrict__ WkB, const float* __restrict__ bkB,
    float* __restrict__ ws)
{
    const int proj = blockIdx.z;
    const float* X; const float* W; const float* b; float* Y;
    switch (proj) {
        case 0:  X = modalA; W = WqA; b = bqA; Y = ws + 0 * PROJ_ELEMS; break;
        case 1:  X = modalA; W = WkA; b = bkA; Y = ws + 1 * PROJ_ELEMS; break;
        case 2:  X = modalB; W = WqB; b = bqB; Y = ws + 2 * PROJ_ELEMS; break;
        default: X = modalB; W = WkB; b = bkB; Y = ws + 3 * PROJ_ELEMS; break;
    }

    const int lane = threadIdx.x & 31;
    const int wave = threadIdx.x >> 5;
    const int l15  = lane & 15;
    const int hi   = lane >> 4;

    const int m0 = blockIdx.x * 64 + wave * 16;   // output row base (16 rows)
    const int c0 = blockIdx.y * 64;               // output col base (64 cols)

    const float* xrow = X + (size_t)(m0 + l15) * EE + 2 * hi;
    const float* wrow[4];
#pragma unroll
    for (int t = 0; t < 4; ++t)
        wrow[t] = W + (size_t)(c0 + 16 * t + l15) * EE + 2 * hi;

    v8f acc[4] = {};

    for (int k = 0; k < EE; k += 4) {
        v2f a = *(const v2f*)(xrow + k);
#pragma unroll
        for (int t = 0; t < 4; ++t) {
            v2f bf = *(const v2f*)(wrow[t] + k);
            acc[t] = __builtin_amdgcn_wmma_f32_16x16x4_f32(
                false, a, false, bf, (short)0, acc[t], false, false);
        }
    }

    // Store with bias. C/D layout: row = vgpr + 8*hi, col = lane&15.
#pragma unroll
    for (int t = 0; t < 4; ++t) {
        const float bias = b[c0 + 16 * t + l15];
#pragma unroll
        for (int v = 0; v < 8; ++v) {
            const int row = m0 + v + 8 * hi;
            Y[(size_t)row * EE + c0 + 16 * t + l15] = acc[t][v] + bias;
        }
    }
}

// ---------------------------------------------------------------------------
// Kernel 2: scores (fp32 WMMA) + row softmax + mean over N, one quadrant tile.
// grid: (1024/16 row tiles, 4 quadrants), block: 256 threads = 8 waves.
// Block owns 16 rows x 1024 cols of one quadrant; wave owns 128 cols
// (8 accumulator tiles). Per n: GEMM -> shfl/LDS row-max -> exp -> row-sum
// -> accumulate probs. Wave-uniform control flow (EXEC all-ones at WMMA).
// Quadrants: 0 = (qA,kA) TL, 1 = (qA,kB) TR, 2 = (qB,kA) BL, 3 = (qB,kB) BR.
// ---------------------------------------------------------------------------
__global__ __launch_bounds__(256) void attn_kernel(
    const float* __restrict__ ws, float* __restrict__ out)
{
    const float* qA = ws + 0 * PROJ_ELEMS;
    const float* kA = ws + 1 * PROJ_ELEMS;
    const float* qB = ws + 2 * PROJ_ELEMS;
    const float* kB = ws + 3 * PROJ_ELEMS;

    const int quad = blockIdx.y;
    const float* Q = (quad < 2) ? qA : qB;
    const float* K = (quad == 0 || quad == 2) ? kA : kB;
    const int rowOff = (quad < 2) ? 0 : LL;
    const int colOff = (quad == 0 || quad == 2) ? 0 : LL;

    const int lane = threadIdx.x & 31;
    const int wave = threadIdx.x >> 5;
    const int l15  = lane & 15;
    const int hi   = lane >> 4;

    const int rowBase = blockIdx.x * 16;   // row tile within quadrant
    const int c0w     = wave * 128;        // this wave's column base

    __shared__ float redmax[8][16];
    __shared__ float redsum[8][16];

    v8f oacc[8] = {};

    for (int n = 0; n < NB; ++n) {
        const float* qbase = Q + ((size_t)n * LL + rowBase + l15) * EE + 2 * hi;
        const float* kbase = K + ((size_t)n * LL + c0w + l15) * EE + 2 * hi;

        // ---- scores GEMM: 16 x 128 strip, K = 512 ----
        v8f sacc[8] = {};
        for (int k = 0; k < EE; k += 4) {
            v2f a = *(const v2f*)(qbase + k);
#pragma unroll
            for (int t = 0; t < 8; ++t) {
                v2f bf = *(const v2f*)(kbase + (size_t)(16 * t) * EE + k);
                sacc[t] = __builtin_amdgcn_wmma_f32_16x16x4_f32(
                    false, a, false, bf, (short)0, sacc[t], false, false);
            }
        }

        // ---- row max: per-vgpr over tiles, butterfly over 16-lane half ----
        float m[8];
#pragma unroll
        for (int v = 0; v < 8; ++v) {
            float mv = sacc[0][v];
#pragma unroll
            for (int t = 1; t < 8; ++t) mv = fmaxf(mv, sacc[t][v]);
            mv = fmaxf(mv, __shfl_xor(mv, 8, 16));
            mv = fmaxf(mv, __shfl_xor(mv, 4, 16));
            mv = fmaxf(mv, __shfl_xor(mv, 2, 16));
            mv = fmaxf(mv, __shfl_xor(mv, 1, 16));
            m[v] = mv;
        }
        if (l15 == 0) {
#pragma unroll
            for (int v = 0; v < 8; ++v) redmax[wave][v + 8 * hi] = m[v];
        }
        __syncthreads();
        float gmax[8];
#pragma unroll
        for (int v = 0; v < 8; ++v) {
            float g = redmax[0][v + 8 * hi];
#pragma unroll
            for (int w = 1; w < 8; ++w) g = fmaxf(g, redmax[w][v + 8 * hi]);
            gmax[v] = g;
        }

        // ---- exp (softmax over pre-scaled scores) + row sum ----
        const float c = INV_SCALE * LOG2E;
        float s[8];
#pragma unroll
        for (int v = 0; v < 8; ++v) {
            float sv = 0.0f;
#pragma unroll
            for (int t = 0; t < 8; ++t) {
                float p = __builtin_amdgcn_exp2f((sacc[t][v] - gmax[v]) * c);
                sacc[t][v] = p;
                sv += p;
            }
            sv += __shfl_xor(sv, 8, 16);
            sv += __shfl_xor(sv, 4, 16);
            sv += __shfl_xor(sv, 2, 16);
            sv += __shfl_xor(sv, 1, 16);
            s[v] = sv;
        }
        if (l15 == 0) {
#pragma unroll
            for (int v = 0; v < 8; ++v) redsum[wave][v + 8 * hi] = s[v];
        }
        __syncthreads();
#pragma unroll
        for (int v = 0; v < 8; ++v) {
            float g = 0.0f;
#pragma unroll
            for (int w = 0; w < 8; ++w) g += redsum[w][v + 8 * hi];
            const float inv = 1.0f / g;
#pragma unroll
            for (int t = 0; t < 8; ++t)
                oacc[t][v] += sacc[t][v] * inv;
        }
        // redmax(n+1) writes are fenced from redmax(n) reads by the second
        // __syncthreads; redsum(n+1) writes fenced by the first. No extra
        // barrier needed.
    }

    // ---- store: mean over N (1/8) ----
#pragma unroll
    for (int t = 0; t < 8; ++t) {
#pragma unroll
        for (int v = 0; v < 8; ++v) {
            const int gr = rowOff + rowBase + v + 8 * hi;
            const int gc = colOff + c0w + 16 * t + l15;
            out[(size_t)gr * (2 * LL) + gc] = oacc[t][v] * 0.125f;
        }
    }
}

// ---------------------------------------------------------------------------
extern "C" void kernel_launch(void* const* d_in, const int* in_sizes, int n_in,
                              void* d_out, int out_size, void* d_ws, size_t ws_size,
                              hipStream_t stream) {
    const float* modalA = (const float*)d_in[0];
    const float* modalB = (const float*)d_in[1];
    const float* WqA = (const float*)d_in[2];
    const float* bqA = (const float*)d_in[3];
    const float* WkA = (const float*)d_in[4];
    const float* bkA = (const float*)d_in[5];
    const float* WqB = (const float*)d_in[6];
    const float* bqB = (const float*)d_in[7];
    const float* WkB = (const float*)d_in[8];
    const float* bkB = (const float*)d_in[9];
    float* ws  = (float*)d_ws;       // 4 x 16 MB: qA, kA, qB, kB
    float* out = (float*)d_out;      // [2048, 2048] f32

    // Stage 1: projections. 8192/64 row blocks, 512/64 col blocks, 4 mats.
    dim3 g1(128, 8, 4);
    proj_kernel<<<g1, 128, 0, stream>>>(modalA, modalB,
                                        WqA, bqA, WkA, bkA,
                                        WqB, bqB, WkB, bkB, ws);

    // Stage 2: scores + softmax + mean. 64 row tiles x 4 quadrants.
    dim3 g2(64, 4);
    attn_kernel<<<g2, 256, 0, stream>>>(ws, out);
}